// GEVFenchelYoungLoss_75076028334876
// MI455X (gfx1250) — compile-verified
//
#include <hip/hip_runtime.h>

typedef __attribute__((ext_vector_type(2))) float v2f;
typedef __attribute__((ext_vector_type(4))) float v4f;
typedef __attribute__((ext_vector_type(8))) float v8f;

#define SQRTPI  1.7724538509055160f
#define LOG2E   1.4426950408889634f
// Entropy of GEV at target=0 / target=1 (xi=0.5, eps=1e-6), precomputed:
//  E = 2*(sqrt(pi)*erfc(sqrt(-log(clip(p,eps,1-eps)))) - clip(p,eps,1-eps))
#define GEV_E0 (-1.4795920e-6f)
#define GEV_E1 (1.5409097f)
#define YLO 1.0000005e-6f      // -log(1-1e-6)
#define YHI 13.815511f         // -log(1e-6)

// dual(x) + ent(target) - target*x, xi = 0.5.
// erfc via Abramowitz-Stegun 7.1.26: erfc(z) = e^{-z^2} * P(1/(1+0.3275911 z)),
// which shares e^{-y} (z^2 == y) with the first Gamma term:
//   Gamma(y,-1/2) = 2 e^{-y} ( rsqrt(y) - sqrt(pi) * P(w) )
__device__ __forceinline__ float gev_loss_elem(float x, float t) {
  float tp = fmaxf(fmaf(0.5f, x, 1.0f), 1e-6f);
  float r  = __builtin_amdgcn_rcpf(tp);
  float y  = fminf(fmaxf(r * r, YLO), YHI);
  float sy = __builtin_amdgcn_sqrtf(y);
  float ry = __builtin_amdgcn_rsqf(y);
  float e  = __builtin_amdgcn_exp2f(y * -LOG2E);          // e^{-y}
  float w  = __builtin_amdgcn_rcpf(fmaf(0.3275911f, sy, 1.0f));
  float p  = fmaf(w, 1.061405429f, -1.453152027f);
  p = fmaf(w, p, 1.421413741f);
  p = fmaf(w, p, -0.284496736f);
  p = fmaf(w, p, 0.254829592f);
  p = p * w;
  float dual = 2.0f * e * fmaf(-SQRTPI, p, ry);
  dual = (x > -2.0f) ? dual : 0.0f;
  return fmaf(t, (GEV_E1 - GEV_E0) - x, dual + GEV_E0);
}

#define BLOCKS 2048
#define TPB 256

__global__ void __launch_bounds__(TPB)
gev_fy_partial(const float* __restrict__ x, const float* __restrict__ tgt,
               float* __restrict__ partial, int n4) {
  const int tid    = blockIdx.x * TPB + threadIdx.x;
  const int stride = BLOCKS * TPB;
  const int full   = n4 / stride;          // uniform, unguarded main loop

  const v4f* __restrict__ x4 = (const v4f*)x;
  const v4f* __restrict__ t4 = (const v4f*)tgt;

  v8f acc = {0.f, 0.f, 0.f, 0.f, 0.f, 0.f, 0.f, 0.f};
  v2f ones; ones.x = 1.0f; ones.y = 1.0f;

  int i = tid;
#pragma unroll 2
  for (int it = 0; it < full; ++it, i += stride) {
    v4f xv = __builtin_nontemporal_load(&x4[i]);
    v4f tv = __builtin_nontemporal_load(&t4[i]);
    v2f a;
    a.x = gev_loss_elem(xv.x, tv.x) + gev_loss_elem(xv.y, tv.y);
    a.y = gev_loss_elem(xv.z, tv.z) + gev_loss_elem(xv.w, tv.w);
    // D = A x ones(4x16) + C : each A value lands in 16 slots -> final /16.
    // EXEC is all-1s here (uniform control flow).
    acc = __builtin_amdgcn_wmma_f32_16x16x4_f32(
        false, a, false, ones, (short)0, acc, false, false);
  }

  // Epilogue: scalar-uniform branch, lane masking via selects (EXEC stays full).
  if (full * stride < n4) {
    bool ok = (i < n4);
    v4f xv = {0.f, 0.f, 0.f, 0.f};
    v4f tv = {0.f, 0.f, 0.f, 0.f};
    if (ok) { xv = x4[i]; tv = t4[i]; }
    float l0 = ok ? gev_loss_elem(xv.x, tv.x) + gev_loss_elem(xv.y, tv.y) : 0.f;
    float l1 = ok ? gev_loss_elem(xv.z, tv.z) + gev_loss_elem(xv.w, tv.w) : 0.f;
    v2f a; a.x = l0; a.y = l1;
    acc = __builtin_amdgcn_wmma_f32_16x16x4_f32(
        false, a, false, ones, (short)0, acc, false, false);
  }

  // Fold this lane's 8 accumulator rows.
  float s = 0.0f;
#pragma unroll
  for (int k = 0; k < 8; ++k) s += acc[k];

  // wave32 tree reduction
#pragma unroll
  for (int off = 16; off > 0; off >>= 1) s += __shfl_down(s, off, 32);

  __shared__ float lds[TPB / 32];
  const int wave = threadIdx.x >> 5;
  const int lane = threadIdx.x & 31;
  if (lane == 0) lds[wave] = s;
  __syncthreads();
  if (threadIdx.x == 0) {
    float b = 0.0f;
#pragma unroll
    for (int w = 0; w < TPB / 32; ++w) b += lds[w];
    partial[blockIdx.x] = b;   // still scaled 16x by the WMMA ones-broadcast
  }
}

__global__ void __launch_bounds__(TPB)
gev_fy_finish(const float* __restrict__ partial, int nparts,
              const float* __restrict__ x, const float* __restrict__ tgt,
              int tail_start, int n, float* __restrict__ out) {
  __shared__ float lds[TPB];
  float s = 0.0f;
  for (int i = threadIdx.x; i < nparts; i += TPB) s += partial[i];
  s *= (1.0f / 16.0f);  // undo WMMA ones-broadcast scaling
  for (int i = tail_start + threadIdx.x; i < n; i += TPB)  // n%4 leftovers
    s += gev_loss_elem(x[i], tgt[i]);
  lds[threadIdx.x] = s;
  __syncthreads();
#pragma unroll
  for (int off = TPB / 2; off > 0; off >>= 1) {
    if (threadIdx.x < off) lds[threadIdx.x] += lds[threadIdx.x + off];
    __syncthreads();
  }
  if (threadIdx.x == 0) out[0] = lds[0] / (float)n;
}

extern "C" void kernel_launch(void* const* d_in, const int* in_sizes, int n_in,
                              void* d_out, int out_size, void* d_ws, size_t ws_size,
                              hipStream_t stream) {
  (void)n_in; (void)out_size; (void)ws_size;
  const float* x   = (const float*)d_in[0];
  const float* tgt = (const float*)d_in[1];
  const int n  = in_sizes[0];       // 8192*8192 = 67,108,864
  const int n4 = n >> 2;            // float4 count
  float* partial = (float*)d_ws;    // BLOCKS floats of scratch

  gev_fy_partial<<<BLOCKS, TPB, 0, stream>>>(x, tgt, partial, n4);
  gev_fy_finish<<<1, TPB, 0, stream>>>(partial, BLOCKS, x, tgt, n4 << 2, n,
                                       (float*)d_out);
}